// Routing_5420248727747
// MI455X (gfx1250) — compile-verified
//
#include <hip/hip_runtime.h>
#include <math.h>

// Problem constants
constexpr int kB   = 64;    // batch
constexpr int kIC  = 4096;  // input capsules
constexpr int kE   = 8;     // input dim (GEMM K)
constexpr int kJ   = 16;    // output capsules
constexpr int kD   = 16;    // output dim
constexpr int kBT  = 16;    // batch tile = WMMA M
constexpr int kIBLK = 64;   // i-range per workgroup
constexpr int kCH  = 8;     // i's per chunk (one per wave, 8 waves)
constexpr int kNBT = kB / kBT;     // 4
constexpr int kNIB = kIC / kIBLK;  // 64

typedef float v2f __attribute__((ext_vector_type(2)));
typedef float v8f __attribute__((ext_vector_type(8)));

__global__ void zero_ws(float* p, int n) {
    int i = blockIdx.x * blockDim.x + threadIdx.x;
    if (i < n) p[i] = 0.0f;
}

// One routing pass: recompute u_hat tiles with fp32 WMMA, do softmax routing in
// LDS, accumulate partial s[b,j,d] via global fp32 atomics.
__global__ void __launch_bounds__(256)
routing_pass(const float* __restrict__ x, const float* __restrict__ W,
             const float* __restrict__ vsumG, float* __restrict__ sG, int pass) {
    extern __shared__ float smem[];
    float* uhat  = smem;                         // [CH][J][BT][D] = 32768 f
    float* c_lds = smem + kCH * kJ * kBT * kD;   // [CH][J][BT]    = 2048 f
    float* vsl   = c_lds + kCH * kJ * kBT;       // [BT][J][D]     = 4096 f

    const int tid = threadIdx.x;
    const int bt  = blockIdx.x & (kNBT - 1);
    const int ib  = blockIdx.x >> 2;
    const int b0  = bt * kBT;
    const int i0  = ib * kIBLK;

    if (pass > 0) {
        for (int k = tid; k < kBT * kJ * kD; k += 256)
            vsl[k] = vsumG[(size_t)b0 * kJ * kD + k];
    }

    const int lane = tid & 31;
    const int wv   = tid >> 5;        // wave id 0..7 -> local i
    const int bl   = lane & 15;       // row/col within 16 (M for A, N for B/D)
    const int half = lane >> 4;       // K-half select
    const int e0   = half * 2;

    float sacc[kBT];
#pragma unroll
    for (int b = 0; b < kBT; ++b) sacc[b] = 0.0f;
    const int jj = tid >> 4;          // 0..15 (s-accum j)
    const int dd = tid & 15;          // 0..15 (s-accum d)

    for (int ch = 0; ch < kIBLK / kCH; ++ch) {
        const int i = i0 + ch * kCH + wv;

        // ---------------- GEMM phase: u_hat tile via V_WMMA_F32_16X16X4_F32 --
        // A[m=b][k=e] = x[b0+m, i, e] ; reused across all 16 j.
        const float* pa = x + ((size_t)(b0 + bl) * kIC + i) * kE;
        v2f a1 = *(const v2f*)(pa + e0);
        v2f a2 = *(const v2f*)(pa + e0 + 4);
#pragma unroll 4
        for (int j = 0; j < kJ; ++j) {
            const float* pb = W + (((size_t)j * kIC + i) * kD + bl) * kE;
            if (j + 1 < kJ)  // uniform branch; prefetch next W tile into caches
                __builtin_prefetch(W + (((size_t)(j + 1) * kIC + i) * kD + bl) * kE, 0, 0);
            v2f b1 = *(const v2f*)(pb + e0);       // B[k=e][n=d] = W[j,i,d,e]
            v2f b2 = *(const v2f*)(pb + e0 + 4);
            v8f acc = {};
            acc = __builtin_amdgcn_wmma_f32_16x16x4_f32(false, a1, false, b1,
                                                        (short)0, acc, false, false);
            acc = __builtin_amdgcn_wmma_f32_16x16x4_f32(false, a2, false, b2,
                                                        (short)0, acc, false, false);
            // D layout: lane holds N=bl (=d), rows M = r + 8*half (=b-local)
            float* uh = uhat + ((size_t)(wv * kJ + j) * kBT) * kD + bl;
#pragma unroll
            for (int r = 0; r < 8; ++r)
                uh[(r + 8 * half) * kD] = acc[r];
        }
        __syncthreads();

        // ---------------- logits + softmax over j (pass > 0) -----------------
        if (pass > 0) {
            if (tid < kBT * kCH) {               // one thread per (b, i) pair
                const int b  = tid & 15;
                const int il = tid >> 4;
                float lg[kJ];
                float mx = -3.4e38f;
#pragma unroll
                for (int j = 0; j < kJ; ++j) {
                    float a = 0.0f;
                    const float* u = uhat + (((size_t)il * kJ + j) * kBT + b) * kD;
                    const float* v = vsl + ((size_t)b * kJ + j) * kD;
#pragma unroll
                    for (int d = 0; d < kD; ++d) a += u[d] * v[d];
                    lg[j] = a;
                    mx = fmaxf(mx, a);
                }
                float s = 0.0f;
#pragma unroll
                for (int j = 0; j < kJ; ++j) { lg[j] = __expf(lg[j] - mx); s += lg[j]; }
                const float inv = 1.0f / s;
#pragma unroll
                for (int j = 0; j < kJ; ++j)
                    c_lds[((size_t)il * kJ + j) * kBT + b] = lg[j] * inv;
            }
            __syncthreads();
        }

        // ---------------- s[b,j,d] partial accumulation ----------------------
        // thread owns (jj, dd); loops b and local i. pass 0: c == 1/16.
#pragma unroll
        for (int b = 0; b < kBT; ++b) {
            float a = sacc[b];
            for (int il = 0; il < kCH; ++il) {
                const float c = (pass == 0) ? 0.0625f
                                            : c_lds[((size_t)il * kJ + jj) * kBT + b];
                a += c * uhat[(((size_t)il * kJ + jj) * kBT + b) * kD + dd];
            }
            sacc[b] = a;
        }
        __syncthreads();   // protect uhat/c before next chunk overwrites
    }

#pragma unroll
    for (int b = 0; b < kBT; ++b)
        atomicAdd(&sG[(((size_t)(b0 + b)) * kJ + jj) * kD + dd], sacc[b]);
}

// squash(s) per (b,j); maintain vsum = Σ v_r; final pass writes d_out.
__global__ void squash_k(const float* __restrict__ sG, float* __restrict__ vsumG,
                         float* __restrict__ out, int pass) {
    int pair = blockIdx.x * blockDim.x + threadIdx.x;   // (b*16 + j), 0..1023
    if (pair >= kB * kJ) return;
    const float* s = sG + (size_t)pair * kD;
    float v[kD];
    float s2 = 0.0f;
#pragma unroll
    for (int d = 0; d < kD; ++d) { v[d] = s[d]; s2 += v[d] * v[d]; }
    const float scale = s2 / (1.0f + s2) * rsqrtf(s2 + 1e-7f);
    if (pass == 0) {
#pragma unroll
        for (int d = 0; d < kD; ++d) vsumG[(size_t)pair * kD + d] = scale * v[d];
    } else if (pass == 1) {
#pragma unroll
        for (int d = 0; d < kD; ++d) vsumG[(size_t)pair * kD + d] += scale * v[d];
    } else {
#pragma unroll
        for (int d = 0; d < kD; ++d) out[(size_t)pair * kD + d] = scale * v[d];
    }
}

extern "C" void kernel_launch(void* const* d_in, const int* in_sizes, int n_in,
                              void* d_out, int out_size, void* d_ws, size_t ws_size,
                              hipStream_t stream) {
    const float* x = (const float*)d_in[0];   // [64, 4096, 8] f32
    const float* W = (const float*)d_in[1];   // [16, 4096, 16, 8] f32
    float* out = (float*)d_out;               // [64, 16, 16] f32

    float* s0   = (float*)d_ws;               // 3 pass-specific s buffers
    float* vsum = s0 + 3 * kB * kJ * kD;      // running Σ v_r

    const int nzero = 3 * kB * kJ * kD;       // s buffers need zero each launch
    zero_ws<<<(nzero + 255) / 256, 256, 0, stream>>>(s0, nzero);

    const size_t shbytes =
        (size_t)(kCH * kJ * kBT * kD + kCH * kJ * kBT + kBT * kJ * kD) * sizeof(float);

    for (int p = 0; p < 3; ++p) {
        float* sp = s0 + (size_t)p * kB * kJ * kD;
        routing_pass<<<kNBT * kNIB, 256, shbytes, stream>>>(x, W, vsum, sp, p);
        squash_k<<<(kB * kJ + 255) / 256, 256, 0, stream>>>(sp, vsum, out, p);
    }
}